// HDRPointwiseNN_73873437491490
// MI455X (gfx1250) — compile-verified
//
#include <hip/hip_runtime.h>
#include <hip/hip_bf16.h>
#include <math.h>

typedef __attribute__((ext_vector_type(2))) float v2f;
typedef __attribute__((ext_vector_type(8))) float v8f;

__device__ __forceinline__ int clampi(int v, int lo, int hi) {
    return v < lo ? lo : (v > hi ? hi : v);
}

// ---------------------------------------------------------------------------
// Branchless im2col operand fetch: unconditional loads of clamped (in-bounds)
// addresses + v_cndmask selects. No EXEC-mask branch ladders, so the compiler
// can batch several global_load_b32 per K-step and hide latency.
// ---------------------------------------------------------------------------
template <int KHW, int KW>
__device__ __forceinline__ void fetch_ab(const float* __restrict__ inb,
                                         const float* __restrict__ wn,
                                         int k, int K, int H, int W,
                                         bool mval, bool nval, int iy0, int ix0,
                                         float& a, float& w)
{
    const int kc = k < K ? k : K - 1;       // clamp K tail
    const int c  = kc / KHW;
    const int r  = kc - c * KHW;
    const int dy = r / KW;
    const int dx = r - dy * KW;
    const int iy = iy0 + dy, ix = ix0 + dx;
    const int iyc = clampi(iy, 0, H - 1);
    const int ixc = clampi(ix, 0, W - 1);
    const bool aok = mval & (k < K) & (iy == iyc) & (ix == ixc);
    const float va = inb[((size_t)c * H + iyc) * W + ixc];  // always in-bounds
    const float vw = wn[kc];                                 // always in-bounds
    a = aok ? va : 0.f;
    w = (nval & (k < K)) ? vw : 0.f;
}

// ---------------------------------------------------------------------------
// Generic implicit-GEMM conv via V_WMMA_F32_16X16X4_F32.
// One wave32 per (batch, 16-pixel M-tile, 16-channel N-tile).
//   A[m][k] = im2col(input),  B[k][n] = weight[n][k],  D = A*B + C
// A layout (32-bit, 16x4): lane L holds M=L%16; VGPR j holds K = k0 + 2*(L/16)+j
// B layout transpose-symmetric; D layout per ISA (8 VGPRs).
// K-loop unrolled x2 (two WMMAs / iter) for memory-level parallelism.
// ---------------------------------------------------------------------------
template <int KHW, int KW>
__global__ __launch_bounds__(32)
void wmma_conv(const float* __restrict__ in, const float* __restrict__ wgt,
               const float* __restrict__ bias, const float* __restrict__ addv,
               float* __restrict__ out,
               int Cin, int H, int W, int Cout, int OH, int OW,
               int stride, int pad, int relu)
{
    const int lane = threadIdx.x;
    const int lm   = lane & 15;
    const int hi   = lane >> 4;            // 0: K pair {0,1}, 1: K pair {2,3}
    const int ntiles = (Cout + 15) >> 4;
    const int M      = OH * OW;
    const int mtiles = (M + 15) >> 4;

    int nt = blockIdx.x % ntiles;
    int t  = blockIdx.x / ntiles;
    int mt = t % mtiles;
    int b  = t / mtiles;

    const int  m    = mt * 16 + lm;
    const int  n    = nt * 16 + lm;
    const bool mval = (m < M);
    const bool nval = (n < Cout);
    const int  mc   = mval ? m : 0;
    const int  oy   = mc / OW;
    const int  ox   = mc - oy * OW;
    const int  iy0  = oy * stride - pad;
    const int  ix0  = ox * stride - pad;

    const float* __restrict__ inb = in + (size_t)b * Cin * H * W;
    const float* __restrict__ wn  = wgt + (size_t)(nval ? n : 0) * Cin * KHW;
    const int K = Cin * KHW;

    v8f acc = {};
    int k0 = 0;
    // main loop: 2 WMMAs per iteration, 8 loads issued before first use
    for (; k0 + 8 <= K; k0 += 8) {
        v2f a0, b0, a1, b1;
#pragma unroll
        for (int j = 0; j < 2; ++j) {
            float a, w;
            fetch_ab<KHW, KW>(inb, wn, k0 + hi * 2 + j, K, H, W,
                              mval, nval, iy0, ix0, a, w);
            a0[j] = a; b0[j] = w;
            fetch_ab<KHW, KW>(inb, wn, k0 + 4 + hi * 2 + j, K, H, W,
                              mval, nval, iy0, ix0, a, w);
            a1[j] = a; b1[j] = w;
        }
        acc = __builtin_amdgcn_wmma_f32_16x16x4_f32(
            false, a0, false, b0, (short)0, acc, false, false);
        acc = __builtin_amdgcn_wmma_f32_16x16x4_f32(
            false, a1, false, b1, (short)0, acc, false, false);
    }
    // tail (K not a multiple of 8, e.g. K=27 for the first splat conv)
    for (; k0 < K; k0 += 4) {
        v2f a0, b0;
#pragma unroll
        for (int j = 0; j < 2; ++j) {
            float a, w;
            fetch_ab<KHW, KW>(inb, wn, k0 + hi * 2 + j, K, H, W,
                              mval, nval, iy0, ix0, a, w);
            a0[j] = a; b0[j] = w;
        }
        acc = __builtin_amdgcn_wmma_f32_16x16x4_f32(
            false, a0, false, b0, (short)0, acc, false, false);
    }

    const float bvadd = (nval && bias) ? bias[n] : 0.f;
    const float gvadd = (nval && addv) ? addv[b * Cout + n] : 0.f;
#pragma unroll
    for (int r = 0; r < 8; ++r) {
        int mm = mt * 16 + r + hi * 8;     // D: VGPR r -> M=r (lanes 0-15) / r+8
        if (mm < M && nval) {
            float v = acc[r] + bvadd + gvadd;
            if (relu) v = fmaxf(v, 0.f);
            int oy2 = mm / OW;
            int ox2 = mm - oy2 * OW;
            out[(((size_t)b * Cout + n) * OH + oy2) * OW + ox2] = v;
        }
    }
}

// ---------------------------------------------------------------------------
// Fused full-res pass: per-pixel guide (1x1 convs + sigmoid), bilateral slice
// from the 96-ch coefficient tensor A (4,96,16,16), affine apply.
// grid[b][gy][gx][d][cc] == A[b][d*12+cc][gy][gx]  (reshape/transpose folded).
// 4 pixels/thread, float4 I/O. ~100 MB HBM traffic total -> ~4.3us roofline.
// ---------------------------------------------------------------------------
__global__ __launch_bounds__(256)
void slice_apply(const float* __restrict__ fullres, const float* __restrict__ A,
                 const float* __restrict__ wgd0, const float* __restrict__ bgd0,
                 const float* __restrict__ wgd1, const float* __restrict__ bgd1,
                 float* __restrict__ out)
{
    __shared__ float sw0[48], sb0[16], sw1[16], sb1[1];
    const int t = threadIdx.x;
    if (t < 48) sw0[t] = wgd0[t];
    if (t < 16) { sb0[t] = bgd0[t]; sw1[t] = wgd1[t]; }
    if (t == 0) sb1[0] = bgd1[0];
    __syncthreads();

    const int tid = blockIdx.x * 256 + t;      // 0 .. 4*1024*1024/4 - 1
    const int b   = tid >> 18;                 // 262144 pixel-quads per image
    const int p   = (tid & 262143) << 2;       // pixel index within image
    const int y   = p >> 10;
    const int x   = p & 1023;

    const float* __restrict__ fb = fullres + (size_t)b * 3 * 1024 * 1024;
    const size_t rowoff = (size_t)y * 1024 + x;
    const float4 r4 = *(const float4*)(fb + rowoff);
    const float4 g4 = *(const float4*)(fb + 1024 * 1024 + rowoff);
    const float4 u4 = *(const float4*)(fb + 2 * 1024 * 1024 + rowoff);
    const float rr[4] = {r4.x, r4.y, r4.z, r4.w};
    const float gg[4] = {g4.x, g4.y, g4.z, g4.w};
    const float uu[4] = {u4.x, u4.y, u4.z, u4.w};

    // i-direction (rows) shared by all 4 pixels
    const float gif = ((float)y + 0.5f) * (1.0f / 64.0f);
    const int   gi0 = (int)floorf(gif - 0.5f);
    const float wi0 = fmaxf(1.f - fabsf((float)gi0 + 0.5f - gif), 0.f);
    const float wi1 = fmaxf(1.f - fabsf((float)gi0 + 1.5f - gif), 0.f);
    const int   giA[2] = {clampi(gi0, 0, 15), clampi(gi0 + 1, 0, 15)};
    const float wiA[2] = {wi0, wi1};

    const float* __restrict__ Ab = A + (size_t)b * 96 * 256;

    float R[4], G[4], Bo[4];
    for (int i = 0; i < 4; ++i) {
        const float r = rr[i], g = gg[i], u = uu[i];

        // guide = sigmoid(wgd1 . relu(wgd0 . rgb + bgd0) + bgd1)
        float acc = sb1[0];
#pragma unroll
        for (int q = 0; q < 16; ++q) {
            float h = sw0[q * 3 + 0] * r + sw0[q * 3 + 1] * g +
                      sw0[q * 3 + 2] * u + sb0[q];
            acc += sw1[q] * fmaxf(h, 0.f);
        }
        const float gval = 1.f / (1.f + __expf(-acc));

        // j-direction (cols)
        const float gjf = ((float)(x + i) + 0.5f) * (1.0f / 64.0f);
        const int   gj0 = (int)floorf(gjf - 0.5f);
        const float wj0 = fmaxf(1.f - fabsf((float)gj0 + 0.5f - gjf), 0.f);
        const float wj1 = fmaxf(1.f - fabsf((float)gj0 + 1.5f - gjf), 0.f);
        const int   gjA[2] = {clampi(gj0, 0, 15), clampi(gj0 + 1, 0, 15)};
        const float wjA[2] = {wj0, wj1};

        // k-direction (guide depth), smoothed lerp
        const float gkf = gval * 8.0f;
        const int   gk0 = (int)floorf(gkf - 0.5f);
        const float d0  = (float)gk0 + 0.5f - gkf;
        const float d1  = (float)gk0 + 1.5f - gkf;
        const float wk0 = fmaxf(1.f - sqrtf(d0 * d0 + 1e-8f), 0.f);
        const float wk1 = fmaxf(1.f - sqrtf(d1 * d1 + 1e-8f), 0.f);
        const int   gkA[2] = {clampi(gk0, 0, 7), clampi(gk0 + 1, 0, 7)};
        const float wkA[2] = {wk0, wk1};

        float coeff[12];
#pragma unroll
        for (int cc = 0; cc < 12; ++cc) coeff[cc] = 0.f;

#pragma unroll
        for (int ii = 0; ii < 2; ++ii)
#pragma unroll
        for (int jj = 0; jj < 2; ++jj) {
            const float wij = wiA[ii] * wjA[jj];
            const int   off = giA[ii] * 16 + gjA[jj];
#pragma unroll
            for (int kk = 0; kk < 2; ++kk) {
                const float w = wij * wkA[kk];
                const float* __restrict__ gp = Ab + (gkA[kk] * 12) * 256 + off;
#pragma unroll
                for (int cc = 0; cc < 12; ++cc)
                    coeff[cc] = fmaf(w, gp[cc * 256], coeff[cc]);
            }
        }
        R[i]  = r * coeff[0] + g * coeff[1] + u * coeff[2] + coeff[9];
        G[i]  = r * coeff[3] + g * coeff[4] + u * coeff[5] + coeff[10];
        Bo[i] = r * coeff[6] + g * coeff[7] + u * coeff[8] + coeff[11];
    }

    float* __restrict__ ob = out + (size_t)b * 3 * 1024 * 1024 + rowoff;
    *(float4*)(ob)                   = make_float4(R[0], R[1], R[2], R[3]);
    *(float4*)(ob + 1024 * 1024)     = make_float4(G[0], G[1], G[2], G[3]);
    *(float4*)(ob + 2 * 1024 * 1024) = make_float4(Bo[0], Bo[1], Bo[2], Bo[3]);
}

static inline int cdiv16(int x) { return (x + 15) / 16; }

extern "C" void kernel_launch(void* const* d_in, const int* in_sizes, int n_in,
                              void* d_out, int out_size, void* d_ws, size_t ws_size,
                              hipStream_t stream)
{
    (void)in_sizes; (void)n_in; (void)out_size; (void)ws_size;
    const float* lowres  = (const float*)d_in[0];
    const float* fullres = (const float*)d_in[1];
    const float* ws0 = (const float*)d_in[2];  const float* bs0 = (const float*)d_in[3];
    const float* ws1 = (const float*)d_in[4];  const float* bs1 = (const float*)d_in[5];
    const float* ws2 = (const float*)d_in[6];  const float* bs2 = (const float*)d_in[7];
    const float* ws3 = (const float*)d_in[8];  const float* bs3 = (const float*)d_in[9];
    const float* wg0 = (const float*)d_in[10]; const float* bg0 = (const float*)d_in[11];
    const float* wg1 = (const float*)d_in[12]; const float* bg1 = (const float*)d_in[13];
    const float* wfc0 = (const float*)d_in[14]; const float* bfc0 = (const float*)d_in[15];
    const float* wfc1 = (const float*)d_in[16]; const float* bfc1 = (const float*)d_in[17];
    const float* wfc2 = (const float*)d_in[18]; const float* bfc2 = (const float*)d_in[19];
    const float* wl0 = (const float*)d_in[20]; const float* bl0 = (const float*)d_in[21];
    const float* wl1 = (const float*)d_in[22];
    const float* wo  = (const float*)d_in[23]; const float* bo  = (const float*)d_in[24];
    const float* wgd0 = (const float*)d_in[25]; const float* bgd0 = (const float*)d_in[26];
    const float* wgd1 = (const float*)d_in[27]; const float* bgd1 = (const float*)d_in[28];
    float* outp = (float*)d_out;

    // bump-allocate scratch (floats, 256B-aligned chunks)
    float* wsf = (float*)d_ws;
    size_t off = 0;
    auto alloc = [&](size_t n) { float* p = wsf + off; off += (n + 63) & ~(size_t)63; return p; };
    float* s0     = alloc((size_t)4 * 8 * 128 * 128);
    float* s1     = alloc((size_t)4 * 16 * 64 * 64);
    float* s2     = alloc((size_t)4 * 32 * 32 * 32);
    float* splat  = alloc((size_t)4 * 64 * 16 * 16);
    float* g0     = alloc((size_t)4 * 64 * 8 * 8);
    float* g1     = alloc((size_t)4 * 64 * 4 * 4);
    float* fc0    = alloc((size_t)4 * 256);
    float* fc1    = alloc((size_t)4 * 128);
    float* glob   = alloc((size_t)4 * 64);
    float* l0     = alloc((size_t)4 * 64 * 16 * 16);
    float* fusion = alloc((size_t)4 * 64 * 16 * 16);
    float* Acoef  = alloc((size_t)4 * 96 * 16 * 16);

    auto conv3 = [&](const float* in, const float* w, const float* bias, const float* addv,
                     float* out, int Cin, int H, int W, int Cout, int OH, int OW,
                     int stride, int pad, int relu) {
        int blocks = 4 * cdiv16(OH * OW) * cdiv16(Cout);
        wmma_conv<9, 3><<<blocks, 32, 0, stream>>>(in, w, bias, addv, out,
            Cin, H, W, Cout, OH, OW, stride, pad, relu);
    };
    auto conv1 = [&](const float* in, const float* w, const float* bias, const float* addv,
                     float* out, int Cin, int H, int W, int Cout, int OH, int OW,
                     int stride, int pad, int relu) {
        int blocks = 4 * cdiv16(OH * OW) * cdiv16(Cout);
        wmma_conv<1, 1><<<blocks, 32, 0, stream>>>(in, w, bias, addv, out,
            Cin, H, W, Cout, OH, OW, stride, pad, relu);
    };

    // splat path
    conv3(lowres, ws0, bs0, nullptr, s0,    3, 256, 256,  8, 128, 128, 2, 1, 1);
    conv3(s0,     ws1, bs1, nullptr, s1,    8, 128, 128, 16,  64,  64, 2, 1, 1);
    conv3(s1,     ws2, bs2, nullptr, s2,   16,  64,  64, 32,  32,  32, 2, 1, 1);
    conv3(s2,     ws3, bs3, nullptr, splat,32,  32,  32, 64,  16,  16, 2, 1, 1);
    // global path
    conv3(splat, wg0, bg0, nullptr, g0, 64, 16, 16, 64, 8, 8, 2, 1, 1);
    conv3(g0,    wg1, bg1, nullptr, g1, 64,  8,  8, 64, 4, 4, 2, 1, 1);
    // FC layers as 1x1 convs on (4, K, 1, 1) — NCHW flatten matches memory order
    conv1(g1,  wfc0, bfc0, nullptr, fc0, 1024, 1, 1, 256, 1, 1, 1, 0, 1);
    conv1(fc0, wfc1, bfc1, nullptr, fc1,  256, 1, 1, 128, 1, 1, 1, 0, 1);
    conv1(fc1, wfc2, bfc2, nullptr, glob, 128, 1, 1,  64, 1, 1, 1, 0, 0);
    // local path; l1 fuses +glob broadcast and ReLU -> fusion directly
    conv3(splat, wl0, bl0, nullptr, l0,     64, 16, 16, 64, 16, 16, 1, 1, 1);
    conv3(l0,    wl1, nullptr, glob, fusion,64, 16, 16, 64, 16, 16, 1, 1, 1);
    // 1x1 projection to 96-channel coefficient tensor
    conv1(fusion, wo, bo, nullptr, Acoef, 64, 16, 16, 96, 16, 16, 1, 0, 0);

    // fused guide + bilateral slice + apply over fullres
    slice_apply<<<4096, 256, 0, stream>>>(fullres, Acoef, wgd0, bgd0, wgd1, bgd1, outp);
}